// Quantization_26697516712523
// MI455X (gfx1250) — compile-verified
//
#include <hip/hip_runtime.h>
#include <hip/hip_bf16.h>

// VQ quantization for MI455X (gfx1250, wave32).
// Bandwidth-bound: ~1.07GB distances stream dominates (floor ~46us @ 23.3TB/s).
// Distance tiles computed with V_WMMA_F32_16X16X4_F32 (full f32 precision):
//   D[i,k] = A[i,:]*B[:,k] + C[i,k],  A[i,0]=x_i, B[0,k]=-2*e_k, C=x_i^2+e_k^2.
// Distances written with non-temporal stores (stream >> 192MB L2).

typedef float v2f __attribute__((ext_vector_type(2)));
typedef float v8f __attribute__((ext_vector_type(8)));

#define NK 512              // codebook entries
#define ROWS_PER_WAVE 16
#define WAVES_PER_BLOCK 8
#define ROWS_PER_BLOCK (ROWS_PER_WAVE * WAVES_PER_BLOCK)  // 128

__global__ __launch_bounds__(256) void vq_wmma_kernel(
    const float* __restrict__ x,     // [N]
    const float* __restrict__ emb,   // [512]
    float* __restrict__ out,         // [N]
    int*   __restrict__ enc_out,     // [N] (int32 bits)
    float* __restrict__ dist,        // [N, 512]
    float* __restrict__ partials)    // [gridDim.x] loss partial sums
{
    __shared__ float s_e[NK];
    __shared__ float s_esq[NK];
    __shared__ float s_wsum[WAVES_PER_BLOCK];

    const int tid = threadIdx.x;
    for (int j = tid; j < NK; j += 256) {
        float e = emb[j];
        s_e[j]   = e;
        s_esq[j] = e * e;
    }
    __syncthreads();

    const int lane = tid & 31;
    const int wave = tid >> 5;
    const int base = blockIdx.x * ROWS_PER_BLOCK + wave * ROWS_PER_WAVE;
    const int half = lane >> 4;        // 0: lanes 0-15, 1: lanes 16-31
    const int hi8  = half * 8;         // C/D rows: M = v + hi8
    const int ln   = lane & 15;        // column within 16-wide tile

    // One coalesced load of this wave's 16 x-values (lanes 0-15 / 16-31 mirror).
    const float xrow = x[base + ln];

    // A matrix (16x4 f32): lanes 0-15 hold K=0 (v0) and K=1 (v1);
    // lanes 16-31 hold K=2/K=3.  Only K=0 row is nonzero: A[i,0] = x_i.
    v2f A;
    A.x = half ? 0.0f : xrow;
    A.y = 0.0f;

    // Per-lane copy of the 8 row values this lane accumulates in C/D
    // (rows base+hi8+v), plus their squares for the C matrix.
    float xv[8], xsq[8];
#pragma unroll
    for (int v = 0; v < 8; ++v) {
        xv[v]  = __shfl(xrow, v + hi8, 32);
        xsq[v] = xv[v] * xv[v];
    }

    float dmin[8];
    int   imin[8];
#pragma unroll
    for (int v = 0; v < 8; ++v) { dmin[v] = 3.4e38f; imin[v] = 0; }

    // 32 tiles of 16 columns cover all 512 codebook entries.
    for (int t = 0; t < 32; ++t) {
        const int col = (t << 4) + ln;

        // B matrix (4x16 f32): only K=0 row nonzero: B[0,n] = -2*e_n.
        const float ecol = s_e[col];
        v2f B;
        B.x = half ? 0.0f : (-2.0f * ecol);
        B.y = 0.0f;

        // C[i,k] = x_i^2 + e_k^2
        const float esq = s_esq[col];
        v8f C;
#pragma unroll
        for (int v = 0; v < 8; ++v) C[v] = xsq[v] + esq;

        // D = A*B + C  -> 16x16 f32 distance tile
        v8f D = __builtin_amdgcn_wmma_f32_16x16x4_f32(
            /*neg_a=*/false, A, /*neg_b=*/false, B,
            /*c_mod=*/(short)0, C, /*reuse_a=*/false, /*reuse_b=*/false);

        // Stream the tile out (non-temporal: 1GB stream > 192MB L2) and
        // update the per-lane running argmin (strict < keeps lowest k).
        size_t rb = (size_t)(base + hi8) * NK + (size_t)col;
#pragma unroll
        for (int v = 0; v < 8; ++v) {
            float dv = D[v];
            __builtin_nontemporal_store(dv, dist + rb + (size_t)v * NK);
            bool lt  = dv < dmin[v];
            dmin[v]  = lt ? dv  : dmin[v];
            imin[v]  = lt ? col : imin[v];
        }
    }

    // Cross-lane argmin over the 16 lanes of each half (each holds 32 of the
    // 512 columns for the same 8 rows).  Tie-break: lower index (jnp.argmin).
#pragma unroll
    for (int v = 0; v < 8; ++v) {
#pragma unroll
        for (int m = 1; m < 16; m <<= 1) {
            float od = __shfl_xor(dmin[v], m, 32);
            int   oi = __shfl_xor(imin[v], m, 32);
            bool take = (od < dmin[v]) || ((od == dmin[v]) && (oi < imin[v]));
            dmin[v] = take ? od : dmin[v];
            imin[v] = take ? oi : imin[v];
        }
    }

    // Lanes 0 and 16 write the 8 rows of their half: cap, lookup, out, loss.
    float lp = 0.0f;
    if (ln == 0) {
#pragma unroll
        for (int v = 0; v < 8; ++v) {
            int row = base + hi8 + v;             // flat index i
            int h   = (row >> 7) & 127;           // H coordinate
            int cap = (h < 4) ? (2 * h + 1) : (NK - 1);
            int e   = imin[v] < cap ? imin[v] : cap;
            float q = s_e[e];
            float d = q - xv[v];
            out[row]     = xv[v] + d;             // straight-through: x+(q-x)
            enc_out[row] = e;
            lp += d * d;
        }
    }

    // Deterministic loss partial: wave shuffle-reduce, then block sum in LDS.
#pragma unroll
    for (int m = 16; m >= 1; m >>= 1) lp += __shfl_xor(lp, m, 32);
    if (lane == 0) s_wsum[wave] = lp;
    __syncthreads();
    if (tid == 0) {
        float s = 0.0f;
        for (int w = 0; w < WAVES_PER_BLOCK; ++w) s += s_wsum[w];
        partials[blockIdx.x] = s;
    }
}

__global__ __launch_bounds__(256) void vq_loss_reduce(
    const float* __restrict__ partials, float* __restrict__ loss_out,
    int nblocks, float scale)
{
    __shared__ float s[256];
    float acc = 0.0f;
    for (int j = threadIdx.x; j < nblocks; j += 256) acc += partials[j];
    s[threadIdx.x] = acc;
    __syncthreads();
    for (int off = 128; off > 0; off >>= 1) {
        if (threadIdx.x < off) s[threadIdx.x] += s[threadIdx.x + off];
        __syncthreads();
    }
    if (threadIdx.x == 0) *loss_out = s[0] * scale;
}

extern "C" void kernel_launch(void* const* d_in, const int* in_sizes, int n_in,
                              void* d_out, int out_size, void* d_ws, size_t ws_size,
                              hipStream_t stream) {
    const float* x   = (const float*)d_in[0];   // [32,1,128,128] f32
    const float* emb = (const float*)d_in[1];   // [512,1] f32
    const int n = in_sizes[0];                  // 524288

    // d_out layout (return-order concat): out[n], encoding[n] (int32),
    // distances[n*512], loss[1]
    float* out   = (float*)d_out;
    int*   enc   = (int*)(out + n);
    float* dist  = out + 2 * (size_t)n;
    float* loss  = dist + (size_t)n * NK;

    float* partials = (float*)d_ws;
    const int blocks = n / ROWS_PER_BLOCK;      // 4096

    vq_wmma_kernel<<<blocks, 256, 0, stream>>>(x, emb, out, enc, dist, partials);
    vq_loss_reduce<<<1, 256, 0, stream>>>(partials, loss, blocks, 2.0f / (float)n);
}